// YatAttention_5377299054792
// MI455X (gfx1250) — compile-verified
//
#include <hip/hip_runtime.h>
#include <hip/hip_bf16.h>
#include <math.h>
#include <stdint.h>

// ---------------------------------------------------------------------------
// Yat attention for MI455X (gfx1250, wave32).
// All matmuls run on V_WMMA_F32_16X16X4_F32 (exact fp32 accumulate, matrix
// pipe). Attention uses flash-style online softmax so the 24 x 2048 x 2048
// score matrices are never materialized.
//
// WMMA fragment layout assumptions (CDNA5 ISA 7.12.2, wave32):
//   A (16x4 f32): lane 0-15 -> M=lane, VGPR0 = K0, VGPR1 = K1;
//                 lane 16-31 -> M=lane-16, VGPR0 = K2, VGPR1 = K3.
//   B (4x16 f32): mirror of A: lane%16 -> N, VGPR0 = K{0|2}, VGPR1 = K{1|3}.
//   C/D (16x16 f32, 8 VGPRs): VGPR r -> M = r + 8*(lane/16), N = lane%16.
//
// LDS layouts are chosen so every WMMA fragment is one aligned 8-byte
// contiguous pair -> single ds_load_b64 into an even-aligned VGPR pair
// (WMMA operands must be even-aligned; avoids v_mov repacking).
// A/K tiles are staged with GLOBAL_LOAD_ASYNC_TO_LDS_B128 (ASYNCcnt).
// ---------------------------------------------------------------------------

typedef float v2f __attribute__((ext_vector_type(2)));
typedef float v8f __attribute__((ext_vector_type(8)));

#define WMMA_F32(a, b, c) \
    __builtin_amdgcn_wmma_f32_16x16x4_f32(false, (a), false, (b), (short)0, (c), false, false)

// Async 16B global -> LDS copy (gfx1250, tracked by ASYNCcnt).
// LDS byte address = low 32 bits of the generic pointer
__device__ __forceinline__ void async_copy_b128(void* lds_dst, const void* gsrc) {
    const uint32_t l = (uint32_t)(uintptr_t)lds_dst;
    const uint64_t g = (uint64_t)(uintptr_t)gsrc;
    asm volatile("global_load_async_to_lds_b128 %0, %1, off"
                 :: "v"(l), "v"(g) : "memory");
}
__device__ __forceinline__ void wait_async0() {
    asm volatile("s_wait_asynccnt 0x0" ::: "memory");
}

// Problem constants (fixed by the harness).
#define BATCH   2
#define SEQ     2048
#define EMB     768
#define HEADS   12
#define HDIM    64
#define ROWS    (BATCH * SEQ)      // 4096
#define QKV_N   (3 * EMB)          // 2304

// ===========================================================================
// Kernel 1/4: tiled fp32 GEMM  C[M,N] = A[M,K] * B[K,N] + bias[N]
// 256 threads = 8 waves; block tile 128x128; wave tile 32x64 (2x4 WMMA tiles)
// ===========================================================================
#define TM 128
#define TN 128
#define BK 16

__global__ __launch_bounds__(256) void gemm_wmma_f32(
    const float* __restrict__ A, const float* __restrict__ B,
    const float* __restrict__ bias, float* __restrict__ C,
    int M, int N, int K)
{
    // A: row-major, stride 20 (even -> b64-aligned pairs; %4==0 -> async b128
    // slots 16B aligned; 20*lane mod 64 spreads banks on column reads).
    __shared__ float As[TM][BK + 4];
    // B: K-pair interleaved: (k, n) -> Bs[k>>1][2*n + (k&1)].
    __shared__ float Bs[BK / 2][2 * TN];

    const int tid  = threadIdx.x;
    const int lane = tid & 31;
    const int wave = tid >> 5;
    const int grp  = lane >> 4;       // 0 or 1 (half-wave)
    const int lr   = lane & 15;

    const int block_m = blockIdx.y * TM;
    const int block_n = blockIdx.x * TN;

    const int wave_m = (wave & 3) * 32;   // 4 waves down M
    const int wave_n = (wave >> 2) * 64;  // 2 waves across N

    v8f acc[2][4] = {};

    for (int k0 = 0; k0 < K; k0 += BK) {
        // ---- stage A tile (128 x 16): async global->LDS b128 ----
        {
            const int r = tid >> 2;            // 0..63
            const int c = (tid & 3) * 4;       // 0,4,8,12
#pragma unroll
            for (int p = 0; p < 2; ++p) {
                const int row = p * 64 + r;
                async_copy_b128(&As[row][c],
                                A + (size_t)(block_m + row) * K + k0 + c);
            }
        }
        // ---- stage B tile (16 x 128), interleaved scatter ----
        {
            const int kr = tid >> 5;           // 0..7 (+8 second pass)
            const int c  = (tid & 31) * 4;     // 0..124
#pragma unroll
            for (int p = 0; p < 2; ++p) {
                const int row = p * 8 + kr;
                const float4 v = *(const float4*)(B + (size_t)(k0 + row) * N + block_n + c);
                float* brow = Bs[row >> 1];
                const int o  = row & 1;
                brow[2 * (c + 0) + o] = v.x;
                brow[2 * (c + 1) + o] = v.y;
                brow[2 * (c + 2) + o] = v.z;
                brow[2 * (c + 3) + o] = v.w;
            }
        }
        wait_async0();
        __syncthreads();

#pragma unroll
        for (int ks = 0; ks < BK; ks += 4) {
            v2f af[2], bf[4];
#pragma unroll
            for (int i = 0; i < 2; ++i) {
                const int row = wave_m + i * 16 + lr;
                af[i] = *(const v2f*)&As[row][ks + 2 * grp];
            }
#pragma unroll
            for (int j = 0; j < 4; ++j) {
                const int col = wave_n + j * 16 + lr;
                bf[j] = *(const v2f*)&Bs[(ks + 2 * grp) >> 1][2 * col];
            }
#pragma unroll
            for (int i = 0; i < 2; ++i)
#pragma unroll
                for (int j = 0; j < 4; ++j)
                    acc[i][j] = WMMA_F32(af[i], bf[j], acc[i][j]);
        }
        __syncthreads();
    }

    // ---- epilogue: C-layout rows r+8*grp, col = lane%16 ----
#pragma unroll
    for (int j = 0; j < 4; ++j) {
        const int col = block_n + wave_n + j * 16 + lr;
        const float bv = bias[col];
#pragma unroll
        for (int i = 0; i < 2; ++i) {
#pragma unroll
            for (int r = 0; r < 8; ++r) {
                const int row = block_m + wave_m + i * 16 + r + 8 * grp;
                C[(size_t)row * N + col] = acc[i][j][r] + bv;
            }
        }
    }
}

// ===========================================================================
// Kernel 2/4: per-(b,h,l) squared norms of q and k rows.
// ===========================================================================
__global__ __launch_bounds__(256) void rowsq_kernel(
    const float* __restrict__ qkv, float* __restrict__ qsq, float* __restrict__ ksq)
{
    const int idx = blockIdx.x * blockDim.x + threadIdx.x;   // over B*H*L
    if (idx >= BATCH * HEADS * SEQ) return;
    const int l  = idx % SEQ;
    const int bh = idx / SEQ;
    const int h  = bh % HEADS;
    const int b  = bh / HEADS;
    const float* q = qkv + ((size_t)(b * SEQ + l)) * QKV_N + h * HDIM;
    const float* k = q + EMB;
    float sq = 0.f, sk = 0.f;
#pragma unroll 8
    for (int d = 0; d < HDIM; ++d) {
        sq = fmaf(q[d], q[d], sq);
        sk = fmaf(k[d], k[d], sk);
    }
    qsq[idx] = sq;
    ksq[idx] = sk;
}

// ===========================================================================
// Kernel 3/4: Yat attention, flash-style. One wave (32 threads) per
// (b, h, 16-query block). Q fragments live in registers for the whole pass.
// ===========================================================================
__global__ __launch_bounds__(32) void yat_attn_kernel(
    const float* __restrict__ qkv, const float* __restrict__ qsq,
    const float* __restrict__ ksq, const int* __restrict__ mask,
    float* __restrict__ out)
{
    // K tile row-major, stride 68 (%4==0: async b128 slots aligned; column
    // reads hit banks 4*lane mod 64 = permutation). Fragment pairs are
    // row-contiguous (even col) -> ds_load_b64.
    __shared__ float Ks[16][68];
    // V tile j-pair interleaved: (j, d) -> Vs[j>>1][2*d + (j&1)].
    __shared__ float Vs[8][128];
    // P tile, stride 18 (even) -> aligned b64 fragment reads.
    __shared__ float Ps[16][18];

    const int itile = blockIdx.x & 127;          // 128 query tiles per (b,h)
    const int bh    = blockIdx.x >> 7;
    const int h     = bh % HEADS;
    const int b     = bh / HEADS;
    const int i0    = itile * 16;

    const int lane = threadIdx.x;
    const int grp  = lane >> 4;
    const int lr   = lane & 15;

    // ---- load Q block (16x64) as 16 A-fragments, kept in registers ----
    const float* qbase = qkv + ((size_t)(b * SEQ + i0)) * QKV_N + h * HDIM;
    v2f aq[16];
#pragma unroll
    for (int ks = 0; ks < 16; ++ks)
        aq[ks] = *(const v2f*)(qbase + (size_t)lr * QKV_N + 4 * ks + 2 * grp);

    // per-row online-softmax state; this lane owns rows r + 8*grp
    float qs[8], mrun[8], lrun[8];
#pragma unroll
    for (int r = 0; r < 8; ++r) {
        qs[r]   = qsq[(size_t)bh * SEQ + i0 + r + 8 * grp];
        mrun[r] = -3.0e38f;
        lrun[r] = 0.0f;
    }
    v8f oacc[4] = {};   // 16 x 64 output accumulator (4 d-tiles)

    const float* kvbase = qkv + ((size_t)b * SEQ) * QKV_N + EMB + h * HDIM;
    const float* ksqrow = ksq + (size_t)bh * SEQ;
    const int*   mrow   = mask + (size_t)b * SEQ;

    for (int j0 = 0; j0 < SEQ; j0 += 16) {
        // ---- stage K tile async, V tile interleaved via VGPRs ----
#pragma unroll
        for (int p = 0; p < 8; ++p) {
            const int li  = p * 128 + lane * 4;   // 0..1023
            const int row = li >> 6;
            const int col = li & 63;
            const float* ksrc = kvbase + (size_t)(j0 + row) * QKV_N + col;
            async_copy_b128(&Ks[row][col], ksrc);
            const float4 v = *(const float4*)(ksrc + EMB);
            float* vrow = Vs[row >> 1];
            const int o = row & 1;
            vrow[2 * (col + 0) + o] = v.x;
            vrow[2 * (col + 1) + o] = v.y;
            vrow[2 * (col + 2) + o] = v.z;
            vrow[2 * (col + 3) + o] = v.w;
        }
        wait_async0();
        __syncthreads();

        // ---- dot tile: Q (16x64) x K^T (64x16) via 16 WMMAs ----
        v8f c = {};
#pragma unroll
        for (int ks = 0; ks < 16; ++ks) {
            const v2f bk = *(const v2f*)&Ks[lr][4 * ks + 2 * grp]; // B[d][j]=K[j][d]
            c = WMMA_F32(aq[ks], bk, c);
        }

        // ---- Yat score + mask + online softmax (row = 16-lane group) ----
        const float kscol = ksqrow[j0 + lr];
        const int   mcol  = mrow[j0 + lr];
#pragma unroll
        for (int r = 0; r < 8; ++r) {
            const float dv = c[r];
            float s = (dv * dv) / (qs[r] + kscol - 2.0f * dv + 1e-6f);
            s = (mcol == 1) ? s : -3.0e38f;

            float tmax = s;
#pragma unroll
            for (int off = 1; off < 16; off <<= 1)
                tmax = fmaxf(tmax, __shfl_xor(tmax, off, 16));

            const float newm  = fmaxf(mrun[r], tmax);
            const float p     = __expf(s - newm);
            const float scale = __expf(mrun[r] - newm);
            mrun[r] = newm;

            float psum = p;
#pragma unroll
            for (int off = 1; off < 16; off <<= 1)
                psum += __shfl_xor(psum, off, 16);
            lrun[r] = lrun[r] * scale + psum;

#pragma unroll
            for (int t = 0; t < 4; ++t)
                oacc[t][r] *= scale;

            Ps[r + 8 * grp][lr] = p;               // C-layout -> LDS
        }
        __syncthreads();

        // ---- P (16x16) x V (16x64) via 16 WMMAs ----
#pragma unroll
        for (int kk = 0; kk < 16; kk += 4) {
            const v2f ap = *(const v2f*)&Ps[lr][kk + 2 * grp];
#pragma unroll
            for (int t = 0; t < 4; ++t) {
                const v2f bv = *(const v2f*)&Vs[(kk + 2 * grp) >> 1][2 * (t * 16 + lr)];
                oacc[t] = WMMA_F32(ap, bv, oacc[t]);
            }
        }
        __syncthreads();
    }

    // ---- normalize and write [B,L,H,D] flattened as rows x 768 ----
#pragma unroll
    for (int t = 0; t < 4; ++t) {
        const int d = t * 16 + lr;
#pragma unroll
        for (int r = 0; r < 8; ++r) {
            const int m = i0 + r + 8 * grp;
            out[((size_t)(b * SEQ + m)) * EMB + h * HDIM + d] = oacc[t][r] / lrun[r];
        }
    }
}

// ===========================================================================
// Launch
// ===========================================================================
extern "C" void kernel_launch(void* const* d_in, const int* in_sizes, int n_in,
                              void* d_out, int out_size, void* d_ws, size_t ws_size,
                              hipStream_t stream)
{
    (void)in_sizes; (void)n_in; (void)out_size; (void)ws_size;

    const float* x    = (const float*)d_in[0];   // [2,2048,768]
    const int*   amsk = (const int*)  d_in[1];   // [2,2048]
    const float* Wqkv = (const float*)d_in[2];   // [768,2304]
    const float* bqkv = (const float*)d_in[3];   // [2304]
    const float* Wout = (const float*)d_in[4];   // [768,768]
    const float* bout = (const float*)d_in[5];   // [768]
    float*       out  = (float*)d_out;           // [4096,768]

    float* qkv  = (float*)d_ws;                          // 4096*2304
    float* attn = qkv  + (size_t)ROWS * QKV_N;           // 4096*768
    float* qsq  = attn + (size_t)ROWS * EMB;             // 49152
    float* ksq  = qsq  + (size_t)BATCH * HEADS * SEQ;    // 49152

    // 1. QKV projection
    dim3 g1(QKV_N / TN, ROWS / TM);
    gemm_wmma_f32<<<g1, 256, 0, stream>>>(x, Wqkv, bqkv, qkv, ROWS, QKV_N, EMB);

    // 2. squared row norms of q, k
    const int nrs = BATCH * HEADS * SEQ;
    rowsq_kernel<<<(nrs + 255) / 256, 256, 0, stream>>>(qkv, qsq, ksq);

    // 3. Yat attention (flash-style)
    yat_attn_kernel<<<BATCH * HEADS * (SEQ / 16), 32, 0, stream>>>(
        qkv, qsq, ksq, amsk, attn);

    // 4. output projection
    dim3 g4(EMB / TN, ROWS / TM);
    gemm_wmma_f32<<<g4, 256, 0, stream>>>(attn, Wout, bout, out, ROWS, EMB, EMB);
}